// MoE_52037823758984
// MI455X (gfx1250) — compile-verified
//
#include <hip/hip_runtime.h>

typedef __attribute__((ext_vector_type(2))) float v2f;
typedef __attribute__((ext_vector_type(8))) float v8f;

#define DFEAT 10

// out[i][j] = b[j] + sum_k x[i][k] * W[j][k], expert picked by route[i].
// WMMA mapping: D(16x16) = A(16x4) * B(4x16) + C, K padded 10 -> 12 (3 chunks).
// Branch-free inner loop: tail tokens (n % 16) handled by a scalar tail kernel.
__global__ __launch_bounds__(256) void moe2_wmma_kernel(
    const float* __restrict__ x,
    const float* __restrict__ W1, const float* __restrict__ b1,
    const float* __restrict__ W2, const float* __restrict__ b2,
    const int*   __restrict__ route,
    float* __restrict__ out, int n_tiles)
{
  const int lane = threadIdx.x & 31;
  const int half = lane >> 4;   // which 16-lane half of the wave
  const int col  = lane & 15;   // N (output feature) column / A row index

  // ---- B matrices (weights), built once per wave ----
  // B[k][n] = W[n][k].  Chunk c, vgpr g, half h -> k = 4c + g + 2h, n = col.
  // Rows k >= 10 are ZERO: this makes A's K=10,11 contents don't-care.
  v2f B1[3], B2[3];
#pragma unroll
  for (int c = 0; c < 3; ++c) {
#pragma unroll
    for (int g = 0; g < 2; ++g) {
      const int k = 4 * c + g + 2 * half;
      float w1v = 0.0f, w2v = 0.0f;
      if (k < DFEAT && col < DFEAT) {
        w1v = W1[col * DFEAT + k];
        w2v = W2[col * DFEAT + k];
      }
      B1[c][g] = w1v;
      B2[c][g] = w2v;
    }
  }
  // Bias: C/D layout has N = col in every accumulator VGPR -> uniform splat.
  const float bv1 = (col < DFEAT) ? b1[col] : 0.0f;
  const float bv2 = (col < DFEAT) ? b2[col] : 0.0f;

  // Loop-invariant per-lane address offsets (elements).
  const int a01off = half * 2;          // A[0]/A[1] base: K = 2*half .. 2*half+1
  const int a2off  = half ? -2 : 8;     // half 0: K=8,9.  half 1: K=10,11 -> B rows
                                        // are zero, so read in-bounds garbage x[row][0,1].

  const int wave    = blockIdx.x * (blockDim.x >> 5) + (threadIdx.x >> 5);
  const int n_waves = gridDim.x * (blockDim.x >> 5);

  for (int t = wave; t < n_tiles; t += n_waves) {
    const int tile = t << 4;

    // ---- A: 16 tokens x K=12.  All v2f loads 8B-aligned, all in-bounds,
    // each of the tile's 640 bytes loaded exactly once (plus 1 dont-care b64).
    const float* xr = x + (tile + col) * DFEAT + a01off;
    const v2f A0 = *(const v2f*)(xr);
    const v2f A1 = *(const v2f*)(xr + 4);
    const v2f A2 = *(const v2f*)(xr + a2off);

    // ---- routes for the 8 output rows this half owns (broadcasts within half) ----
    const int rbase = tile + 8 * half;
    int rt[8];
#pragma unroll
    for (int g = 0; g < 8; ++g) rt[g] = route[rbase + g];

    // ---- both experts via V_WMMA_F32_16X16X4_F32, C seeded with bias ----
    v8f acc1, acc2;
#pragma unroll
    for (int g = 0; g < 8; ++g) { acc1[g] = bv1; acc2[g] = bv2; }

    acc1 = __builtin_amdgcn_wmma_f32_16x16x4_f32(false, A0, false, B1[0], (short)0, acc1, false, false);
    acc2 = __builtin_amdgcn_wmma_f32_16x16x4_f32(false, A0, false, B2[0], (short)0, acc2, false, false);
    acc1 = __builtin_amdgcn_wmma_f32_16x16x4_f32(false, A1, false, B1[1], (short)0, acc1, false, false);
    acc2 = __builtin_amdgcn_wmma_f32_16x16x4_f32(false, A1, false, B2[1], (short)0, acc2, false, false);
    acc1 = __builtin_amdgcn_wmma_f32_16x16x4_f32(false, A2, false, B1[2], (short)0, acc1, false, false);
    acc2 = __builtin_amdgcn_wmma_f32_16x16x4_f32(false, A2, false, B2[2], (short)0, acc2, false, false);

    // ---- select per row and store the 10 valid columns ----
    // C VGPR g, lane l: row = 8*half + g, col = lane&15.  Single exec region.
    if (col < DFEAT) {
#pragma unroll
      for (int g = 0; g < 8; ++g) {
        out[(rbase + g) * DFEAT + col] = (rt[g] == 0) ? acc1[g] : acc2[g];
      }
    }
  }
}

// Scalar tail for n_tokens % 16 (not hit for N = 2^21, kept for generality).
__global__ void moe2_tail_kernel(
    const float* __restrict__ x,
    const float* __restrict__ W1, const float* __restrict__ b1,
    const float* __restrict__ W2, const float* __restrict__ b2,
    const int*   __restrict__ route,
    float* __restrict__ out, int tok0, int n_tokens)
{
  const int idx = tok0 * DFEAT + blockIdx.x * blockDim.x + threadIdx.x;
  if (idx >= n_tokens * DFEAT) return;
  const int i = idx / DFEAT, j = idx % DFEAT;
  const float* W = (route[i] == 0) ? W1 : W2;
  const float* b = (route[i] == 0) ? b1 : b2;
  float s = b[j];
#pragma unroll
  for (int k = 0; k < DFEAT; ++k) s += x[i * DFEAT + k] * W[j * DFEAT + k];
  out[idx] = s;
}

extern "C" void kernel_launch(void* const* d_in, const int* in_sizes, int n_in,
                              void* d_out, int out_size, void* d_ws, size_t ws_size,
                              hipStream_t stream) {
  // setup_inputs order: x, W1, b1, W2, b2, route
  const float* x     = (const float*)d_in[0];
  const float* W1    = (const float*)d_in[1];
  const float* b1    = (const float*)d_in[2];
  const float* W2    = (const float*)d_in[3];
  const float* b2    = (const float*)d_in[4];
  const int*   route = (const int*)d_in[5];
  float* out = (float*)d_out;

  const int n_tokens = in_sizes[0] / DFEAT;
  const int n_tiles  = n_tokens >> 4;      // full 16-token tiles

  if (n_tiles > 0) {
    const int block = 256;                 // 8 wave32 waves per block
    const int waves_per_block = block / 32;
    int grid = (n_tiles + waves_per_block - 1) / waves_per_block;
    if (grid > 2048) grid = 2048;          // grid-stride: long per-wave streams
    moe2_wmma_kernel<<<grid, block, 0, stream>>>(x, W1, b1, W2, b2, route, out,
                                                 n_tiles);
  }

  const int rem = n_tokens & 15;
  if (rem > 0) {
    const int tok0 = n_tiles << 4;
    const int work = rem * DFEAT;
    moe2_tail_kernel<<<(work + 127) / 128, 128, 0, stream>>>(
        x, W1, b1, W2, b2, route, out, tok0, n_tokens);
  }
}